// CannyFilter_2654289789547
// MI455X (gfx1250) — compile-verified
//
#include <hip/hip_runtime.h>
#include <math.h>

// ---- fixed problem geometry (from reference setup_inputs) ----
#define IMG_B 16
#define IMG_C 3
#define IMG_H 512
#define IMG_W 512

#define TILE 64           // output tile per block
#define MR   70           // rows in mean/raw region = TILE + 2*3 halo
#define STR  72           // LDS row stride AND loaded column count (16B-vec friendly)
#define BR   68           // blurred region = TILE + 2*2
#define GR   66           // mag region     = TILE + 2*1

// LDS layout (floats), buffers reused in-place across barriers:
//   t0: raw ch0 @0, ch1 @5040, ch2 @10080   (async loads, 70 rows x 72 cols)
//   t1: mean    @0 (in-place over ch0)
//   t2: htmp    @5040 (over dead ch1)
//   t3: blur    @10080 (over dead ch2)
//   t4: mag     @0 (over dead mean), posb bytes @19008 (just past mag: dead mean tail + htmp)
#define HT   5040
#define BLU  10080
#define POSB 19008        // byte offset of direction-index array

// separable Gaussian: gv = [b,1,b]/(1+2b), b = exp(-0.5)
#define GE 0.27406862f
#define GC 0.45186276f
// orientation bin thresholds: tan(pi/16), tan(3pi/16), tan(5pi/16), tan(7pi/16)
#define TH1 0.19891237f
#define TH2 0.66817864f
#define TH3 1.49660576f
#define TH4 5.02733949f

__global__ __launch_bounds__(256) void canny_fused_kernel(
    const float* __restrict__ img, float* __restrict__ out) {
  __shared__ __align__(16) float lds[15120];  // 60,480 bytes static LDS

  const int tid = threadIdx.x;
  const int b   = blockIdx.x >> 6;           // 64 tiles per image (8x8)
  const int tt  = blockIdx.x & 63;
  const int gy0 = (tt >> 3) << 6;
  const int gx0 = (tt & 7) << 6;
  const float* imgB = img + (size_t)b * (IMG_C * IMG_H * IMG_W);
  const unsigned ldsBase = (unsigned)(size_t)(const void*)&lds[0];

  // Loaded window: rows gy0-3+r (r<70, row-clamped), cols gx0-4+c (c<72).
  if (gx0 != 0 && gx0 != (IMG_W - TILE)) {
    // ---- column-interior tile: wide async b128 loads (16 B/lane/issue) ----
    for (int p = tid; p < MR * (STR / 4) * IMG_C; p += 256) {   // 3780
      const int ch = p / (MR * (STR / 4));
      const int q  = p - ch * (MR * (STR / 4));
      const int r  = q / (STR / 4);
      const int cq = q - r * (STR / 4);                 // vec4 index within row
      const int iy = min(max(gy0 - 3 + r, 0), IMG_H - 1);
      const unsigned long long g =
          (unsigned long long)(size_t)(imgB + (size_t)ch * (IMG_H * IMG_W) +
                                       (size_t)iy * IMG_W + (gx0 - 4) + 4 * cq);
      const unsigned l = ldsBase + (unsigned)(ch * HT + r * STR + 4 * cq) * 4u;
      asm volatile("global_load_async_to_lds_b128 %0, %1, off"
                   :: "v"(l), "v"(g) : "memory");
    }
  } else {
    // ---- edge-column tile: per-lane clamped b32 loads == replication pad #1 ----
    for (int p = tid; p < MR * STR; p += 256) {                 // 5040
      const int r = p / STR, c = p - r * STR;
      const int iy = min(max(gy0 - 3 + r, 0), IMG_H - 1);
      const int ix = min(max(gx0 - 4 + c, 0), IMG_W - 1);
      const unsigned long long g0 =
          (unsigned long long)(size_t)(imgB + (size_t)iy * IMG_W + ix);
      const unsigned l = ldsBase + (unsigned)p * 4u;
      asm volatile("global_load_async_to_lds_b32 %0, %1, off"
                   :: "v"(l), "v"(g0) : "memory");
      asm volatile("global_load_async_to_lds_b32 %0, %1, off"
                   :: "v"(l + HT * 4u), "v"(g0 + 1048576ull) : "memory");
      asm volatile("global_load_async_to_lds_b32 %0, %1, off"
                   :: "v"(l + BLU * 4u), "v"(g0 + 2097152ull) : "memory");
    }
  }
  asm volatile("s_wait_asynccnt 0" ::: "memory");
  __syncthreads();

  // ---- stage 1: channel mean in place, flat vec4 (folds the /C of Sobel) ----
  {
    float4* m4 = (float4*)lds;
    const float4* c1 = (const float4*)&lds[HT];
    const float4* c2 = (const float4*)&lds[BLU];
    for (int i = tid; i < (MR * STR) / 4; i += 256) {           // 1260
      float4 a = m4[i], u = c1[i], w = c2[i];
      a.x = (a.x + u.x + w.x) * (1.0f / 3.0f);
      a.y = (a.y + u.y + w.y) * (1.0f / 3.0f);
      a.z = (a.z + u.z + w.z) * (1.0f / 3.0f);
      a.w = (a.w + u.w + w.w) * (1.0f / 3.0f);
      m4[i] = a;
    }
  }
  __syncthreads();

  // ---- stage 2: horizontal 3-tap Gaussian (70 rows x 68 cols) ----
  // htmp col c is centered at image col gx0-2+c  ->  mean col c+2.
  for (int p = tid; p < MR * BR; p += 256) {
    const int r = p / BR, c = p - r * BR;
    const float* m = &lds[r * STR + c + 1];
    lds[HT + r * STR + c] = GE * (m[0] + m[2]) + GC * m[1];
  }
  __syncthreads();

  // ---- stage 3: vertical 3-tap Gaussian -> blurred (68 x 68) ----
  for (int p = tid; p < BR * BR; p += 256) {
    const int r = p / BR, c = p - r * BR;
    const float* h = &lds[HT + r * STR + c];
    lds[BLU + r * STR + c] = GE * (h[0] + h[2 * STR]) + GC * h[STR];
  }
  __syncthreads();

  // ---- stage 4: Sobel + magnitude + orientation bin (66 x 66) ----
  // BLUR_AT clamps at the *blurred* level == reference's 'edge' pad #2.
  unsigned char* posb = (unsigned char*)lds + POSB;
  for (int p = tid; p < GR * GR; p += 256) {
    const int r = p / GR, c = p - r * GR;
    const int y = gy0 - 1 + r, x = gx0 - 1 + c;

    #define BLUR_AT(Y, X) \
      lds[BLU + (min(max((Y), 0), IMG_H - 1) - (gy0 - 2)) * STR \
              + (min(max((X), 0), IMG_W - 1) - (gx0 - 2))]
    const float tl = BLUR_AT(y - 1, x - 1), tc = BLUR_AT(y - 1, x), tr = BLUR_AT(y - 1, x + 1);
    const float ml = BLUR_AT(y,     x - 1),                         mr = BLUR_AT(y,     x + 1);
    const float bl = BLUR_AT(y + 1, x - 1), bc = BLUR_AT(y + 1, x), br = BLUR_AT(y + 1, x + 1);
    #undef BLUR_AT

    const float gxv = 0.5f * (tr - tl) + (mr - ml) + 0.5f * (br - bl);
    const float gyv = 0.5f * (bl - tl) + (bc - tc) + 0.5f * (br - tr);
    const float mag = sqrtf(gxv * gxv + gyv * gyv);
    const bool inimg = ((unsigned)y < (unsigned)IMG_H) && ((unsigned)x < (unsigned)IMG_W);
    lds[r * STR + c] = inimg ? mag : 0.0f;   // zero outside image == conv zero-pad

    // pos_idx = round(atan(gy/gx)*8/pi + 4) % 8, folded mod 4 via tan thresholds
    const float t = gyv / gxv;
    const float a = fabsf(t);
    const int k = (a < TH1) ? 0 : (a < TH2) ? 1 : (a < TH3) ? 2 : (a < TH4) ? 3 : 4;
    const int j = (t >= 0.0f) ? (k & 3) : ((4 - k) & 3);
    posb[r * STR + c] = (unsigned char)j;
  }
  __syncthreads();

  // ---- stage 5: directional NMS + vec4 store (64 rows x 16 col-groups) ----
  for (int p = tid; p < TILE * (TILE / 4); p += 256) {          // 1024
    const int r = p >> 4, cg = p & 15;
    const int c0 = cg << 2;
    const int y = gy0 + r;

    float res[4];
    #pragma unroll
    for (int u = 0; u < 4; ++u) {
      const int c = c0 + u;
      const int x = gx0 + c;
      const int rr = r + 1, cc = c + 1;

      const float m = lds[rr * STR + cc];
      const int   j = posb[rr * STR + cc];
      // j -> (dy,dx): 0:(0,1) 1:(-1,1) 2:(-1,0) 3:(-1,-1)
      const int dy = (j == 0) ? 0 : -1;
      const int dx = (j == 3) ? -1 : ((j == 2) ? 0 : 1);

      #define MAG_AT(Y, X) \
        ((((unsigned)(Y) < (unsigned)IMG_H) && ((unsigned)(X) < (unsigned)IMG_W)) \
          ? lds[((Y) - (gy0 - 1)) * STR + ((X) - (gx0 - 1))] : 0.0f)
      const float n1 = MAG_AT(y + dy, x + dx);
      const float n2 = MAG_AT(y - dy, x - dx);
      #undef MAG_AT

      res[u] = (m > n1 && m > n2) ? m : 0.0f;
    }
    float4 v; v.x = res[0]; v.y = res[1]; v.z = res[2]; v.w = res[3];
    *(float4*)&out[((size_t)b * IMG_H + y) * IMG_W + gx0 + c0] = v;
  }
}

extern "C" void kernel_launch(void* const* d_in, const int* in_sizes, int n_in,
                              void* d_out, int out_size, void* d_ws, size_t ws_size,
                              hipStream_t stream) {
  (void)in_sizes; (void)n_in; (void)d_ws; (void)ws_size; (void)out_size;
  const float* img = (const float*)d_in[0];   // [16,3,512,512] fp32
  float* out = (float*)d_out;                 // [16,1,512,512] fp32
  const int blocks = IMG_B * (IMG_H / TILE) * (IMG_W / TILE);  // 1024
  canny_fused_kernel<<<blocks, 256, 0, stream>>>(img, out);
}